// tPatchTraj_7078106104301
// MI455X (gfx1250) — compile-verified
//
#include <hip/hip_runtime.h>
#include <hip/hip_bf16.h>
#include <cstdint>
#include <cstddef>

typedef __attribute__((ext_vector_type(2))) float v2f;
typedef __attribute__((ext_vector_type(8))) float v8f;

#define EPSF 1e-5f

// ---------------------------------------------------------------------------
// KNN scan: 16 sequential steps; carried anchor points in `cur`.
// ---------------------------------------------------------------------------
__global__ __launch_bounds__(256) void init_cur_kernel(const float* __restrict__ pts,
                                                       float* __restrict__ cur) {
  int idx = blockIdx.x * 256 + threadIdx.x;
  if (idx < 2 * 1024 * 3) {
    int b = idx / 3072;
    int r = idx % 3072;
    cur[idx] = pts[(size_t)b * (16 * 1024 * 3) + r];  // point_seq[b, 0, :, :]
  }
}

__global__ __launch_bounds__(256) void knn_step_kernel(const float* __restrict__ pts,
                                                       float* __restrict__ cur,
                                                       float* __restrict__ pp,
                                                       float* __restrict__ pf,
                                                       int t) {
  __shared__ float cand[256 * 3];
  const int b = blockIdx.y;
  const int n = blockIdx.x * 256 + threadIdx.x;
  const float qx = cur[(b * 1024 + n) * 3 + 0];
  const float qy = cur[(b * 1024 + n) * 3 + 1];
  const float qz = cur[(b * 1024 + n) * 3 + 2];

  float bd[16];
  int bi[16];
#pragma unroll
  for (int i = 0; i < 16; ++i) { bd[i] = 3.4e38f; bi[i] = 0; }

  const float* frame = pts + (size_t)(b * 16 + t) * (1024 * 3);
  for (int jc = 0; jc < 1024; jc += 256) {
    __syncthreads();
    for (int idx = threadIdx.x; idx < 256 * 3; idx += 256) cand[idx] = frame[jc * 3 + idx];
    __syncthreads();
    for (int jj = 0; jj < 256; ++jj) {
      const float dx = qx - cand[jj * 3 + 0];
      const float dy = qy - cand[jj * 3 + 1];
      const float dz = qz - cand[jj * 3 + 2];
      const float d2 = dx * dx + dy * dy + dz * dz;
      if (d2 < bd[15]) {  // branch-free sorted insertion (stable: ties keep lower index)
        const int cid = jc + jj;
#pragma unroll
        for (int m = 15; m >= 0; --m) {
          const bool up = (m > 0) && (d2 < bd[m - 1]);
          if (up) { bd[m] = bd[m - 1]; bi[m] = bi[m - 1]; }
          else if (d2 < bd[m]) { bd[m] = d2; bi[m] = cid; }
        }
      }
    }
  }

  const int tp = (t > 0) ? (t - 1) : 0;
  const float* prev = pts + (size_t)(b * 16 + tp) * (1024 * 3);
  const int base = ((b * 16 + t) * 1024 + n) * 16;
#pragma unroll
  for (int kk = 0; kk < 16; ++kk) {
    const int id = bi[kk];
#pragma unroll
    for (int d = 0; d < 3; ++d) {
      pp[(size_t)(base + kk) * 3 + d] = frame[id * 3 + d];
      pf[(size_t)(base + kk) * 3 + d] = prev[id * 3 + d];
    }
  }
#pragma unroll
  for (int d = 0; d < 3; ++d) cur[(b * 1024 + n) * 3 + d] = frame[bi[0] * 3 + d];
}

// ---------------------------------------------------------------------------
// Layer 0: features (pf, pp - anchor) -> 6x64 GEMM + bias. Deterministic stats.
// ---------------------------------------------------------------------------
__global__ __launch_bounds__(256) void layer0_kernel(const float* __restrict__ pp,
                                                     const float* __restrict__ pf,
                                                     const float* __restrict__ w0,
                                                     const float* __restrict__ b0,
                                                     float* __restrict__ z0,
                                                     float* __restrict__ statsP) {
  __shared__ float feat[256 * 6];
  __shared__ float Wl[64 * 6];
  __shared__ float red[4 * 64 * 2];
  const int tid = threadIdx.x;
  const int sb = blockIdx.x * 256;
  if (tid < 64 * 6) Wl[tid] = w0[tid];

  const int s = sb + tid;
  const int n = (s >> 4) & 1023;
  const int b = s >> 18;
  const int abase = (b * 262144 + n * 16) * 3;  // pp[b, t=0, n, k=0, :]
#pragma unroll
  for (int d = 0; d < 3; ++d) {
    feat[tid * 6 + d] = pf[(size_t)s * 3 + d];
    feat[tid * 6 + 3 + d] = pp[(size_t)s * 3 + d] - pp[abase + d];
  }
  __syncthreads();

  const int ch = tid & 63;
  const int stripe = tid >> 6;
  float wr[6];
#pragma unroll
  for (int c = 0; c < 6; ++c) wr[c] = Wl[ch * 6 + c];
  const float bb = b0[ch];
  float sum = 0.f, sq = 0.f;
  for (int i = 0; i < 64; ++i) {
    const int sl = stripe * 64 + i;
    float z = bb;
#pragma unroll
    for (int c = 0; c < 6; ++c) z = fmaf(wr[c], feat[sl * 6 + c], z);
    z0[(size_t)(sb + sl) * 64 + ch] = z;
    sum += z;
    sq += z * z;
  }
  red[(stripe * 64 + ch) * 2 + 0] = sum;
  red[(stripe * 64 + ch) * 2 + 1] = sq;
  __syncthreads();
  if (tid < 64) {
    float s4 = 0.f, q4 = 0.f;
#pragma unroll
    for (int st = 0; st < 4; ++st) {
      s4 += red[(st * 64 + tid) * 2 + 0];
      q4 += red[(st * 64 + tid) * 2 + 1];
    }
    statsP[((size_t)blockIdx.x * 64 + tid) * 2 + 0] = s4;
    statsP[((size_t)blockIdx.x * 64 + tid) * 2 + 1] = q4;
  }
}

// ---------------------------------------------------------------------------
// Per-channel finalize: mean/var -> scale/shift. One block per channel.
// ---------------------------------------------------------------------------
__global__ __launch_bounds__(256) void finalize_stats_kernel(const float* __restrict__ partial,
                                                             int nblocks, int C, float invCount,
                                                             const float* __restrict__ g,
                                                             const float* __restrict__ be,
                                                             float* __restrict__ scale,
                                                             float* __restrict__ shift) {
  __shared__ float ss[256], qq[256];
  const int ch = blockIdx.x;
  const int tid = threadIdx.x;
  float s = 0.f, q = 0.f;
  for (int i = tid; i < nblocks; i += 256) {
    s += partial[((size_t)i * C + ch) * 2 + 0];
    q += partial[((size_t)i * C + ch) * 2 + 1];
  }
  ss[tid] = s;
  qq[tid] = q;
  __syncthreads();
  for (int off = 128; off > 0; off >>= 1) {
    if (tid < off) { ss[tid] += ss[tid + off]; qq[tid] += qq[tid + off]; }
    __syncthreads();
  }
  if (tid == 0) {
    const float mean = ss[0] * invCount;
    const float var = qq[0] * invCount - mean * mean;
    const float sc = g[ch] * rsqrtf(var + EPSF);
    scale[ch] = sc;
    shift[ch] = be[ch] - mean * sc;
  }
}

// ---------------------------------------------------------------------------
// WMMA GEMM: [M,K] x [K,N], BN+ReLU applied to input on load.
// Block = 256 threads (8 waves), tile M=128, full N, K chunked by 32 via LDS.
// One wave's 16 rows == one k-pooling group (POOL mode emits group max/min).
// ---------------------------------------------------------------------------
template <int K, int N, bool POOL>
__global__ __launch_bounds__(256) void gemm_bn_kernel(const float* __restrict__ zin,
                                                      const float* __restrict__ scale,
                                                      const float* __restrict__ shift,
                                                      const float* __restrict__ W,     // [N][K]
                                                      const float* __restrict__ bias,  // [N]
                                                      float* __restrict__ zout,
                                                      float* __restrict__ gmax,
                                                      float* __restrict__ gmin,
                                                      float* __restrict__ statsP) {
  constexpr int NT = N / 16;
  __shared__ float lds[4096 + 32 * N];
  float* At = lds;
  float* Bt = lds + 4096;
  const int tid = threadIdx.x;
  const int wave = tid >> 5;
  const int lane = tid & 31;
  const int j = lane & 15;
  const int h = lane >> 4;
  const int row0 = blockIdx.x * 128;

  v8f acc[NT];
  const v8f vzero = {0.f, 0.f, 0.f, 0.f, 0.f, 0.f, 0.f, 0.f};
#pragma unroll
  for (int i = 0; i < NT; ++i) acc[i] = vzero;

  for (int kc = 0; kc < K; kc += 32) {
    __syncthreads();
    // A tile 128x32, fused BN+ReLU of previous layer's pre-activation.
    for (int idx = tid; idx < 4096; idx += 256) {
      const int r = idx >> 5, c = idx & 31;
      const float z = zin[(size_t)(row0 + r) * K + kc + c];
      At[idx] = fmaxf(fmaf(z, scale[kc + c], shift[kc + c]), 0.f);
    }
    // B tile 32xN, stored as (k-even,k-odd) pairs for float2 fragment loads.
    for (int idx = tid; idx < 32 * N; idx += 256) {
      const int kx = idx / N, n = idx % N;
      Bt[(((kx >> 1) * N) + n) * 2 + (kx & 1)] = W[(size_t)n * K + kc + kx];
    }
    __syncthreads();
#pragma unroll
    for (int kx = 0; kx < 32; kx += 4) {
      const v2f a = *(const v2f*)&At[(wave * 16 + j) * 32 + kx + 2 * h];
      const int p = (kx + 2 * h) >> 1;
#pragma unroll
      for (int nt = 0; nt < NT; ++nt) {
        const v2f bf = *(const v2f*)&Bt[(p * N + nt * 16 + j) * 2];
        acc[nt] = __builtin_amdgcn_wmma_f32_16x16x4_f32(false, a, false, bf, (short)0,
                                                        acc[nt], false, false);
      }
    }
  }
  __syncthreads();

  float* Wst = lds;  // reuse A region: 8 waves * N channels * {sum,sumsq}
#pragma unroll
  for (int nt = 0; nt < NT; ++nt) {
    const int ch = nt * 16 + j;  // channel is constant per lane within an n-tile
    const float bv = bias[ch];
    float s = 0.f, q = 0.f;
    if (POOL) {
      float mx = -3.4e38f, mn = 3.4e38f;
#pragma unroll
      for (int v = 0; v < 8; ++v) {
        const float z = acc[nt][v] + bv;
        mx = fmaxf(mx, z);
        mn = fminf(mn, z);
        s += z;
        q += z * z;
      }
      mx = fmaxf(mx, __shfl_xor(mx, 16, 32));
      mn = fminf(mn, __shfl_xor(mn, 16, 32));
      s += __shfl_xor(s, 16, 32);
      q += __shfl_xor(q, 16, 32);
      if (h == 0) {
        const int grp = blockIdx.x * 8 + wave;  // 16 rows = one (b,t,n) group
        gmax[(size_t)grp * N + ch] = mx;
        gmin[(size_t)grp * N + ch] = mn;
        Wst[(wave * N + ch) * 2 + 0] = s;
        Wst[(wave * N + ch) * 2 + 1] = q;
      }
    } else {
#pragma unroll
      for (int v = 0; v < 8; ++v) {
        const float z = acc[nt][v] + bv;  // rows: M = v (+8 for upper half)
        zout[(size_t)(row0 + wave * 16 + h * 8 + v) * N + ch] = z;
        s += z;
        q += z * z;
      }
      s += __shfl_xor(s, 16, 32);
      q += __shfl_xor(q, 16, 32);
      if (h == 0) {
        Wst[(wave * N + ch) * 2 + 0] = s;
        Wst[(wave * N + ch) * 2 + 1] = q;
      }
    }
  }
  __syncthreads();
  for (int ch = tid; ch < N; ch += 256) {
    float s = 0.f, q = 0.f;
#pragma unroll
    for (int w = 0; w < 8; ++w) {
      s += Wst[(w * N + ch) * 2 + 0];
      q += Wst[(w * N + ch) * 2 + 1];
    }
    statsP[((size_t)blockIdx.x * N + ch) * 2 + 0] = s;
    statsP[((size_t)blockIdx.x * N + ch) * 2 + 1] = q;
  }
}

// ---------------------------------------------------------------------------
// max-over-k + BN2 + ReLU: max(relu(z*sc+sh)) = relu((sc>=0?zmax:zmin)*sc+sh)
// ---------------------------------------------------------------------------
__global__ __launch_bounds__(256) void pool_bn_kernel(const float* __restrict__ gmax,
                                                      const float* __restrict__ gmin,
                                                      const float* __restrict__ sc2,
                                                      const float* __restrict__ sh2,
                                                      float* __restrict__ xp) {
  const int i = blockIdx.x * 256 + threadIdx.x;
  const int ch = i & 255;
  const float sc = sc2[ch], sh = sh2[ch];
  const float v = (sc >= 0.f) ? gmax[i] : gmin[i];
  xp[i] = fmaxf(fmaf(v, sc, sh), 0.f);
}

// ---------------------------------------------------------------------------
// Temporal conv as WMMA GEMM: K = 3*256 over shifted time windows.
// ---------------------------------------------------------------------------
__global__ __launch_bounds__(256) void tconv_kernel(const float* __restrict__ xp,
                                                    const float* __restrict__ wt,  // [256][256][3]
                                                    const float* __restrict__ bt,
                                                    float* __restrict__ y,
                                                    float* __restrict__ statsP) {
  constexpr int N = 256;
  constexpr int NT = 16;
  __shared__ float lds[4096 + 32 * N];
  float* At = lds;
  float* Bt = lds + 4096;
  const int tid = threadIdx.x;
  const int wave = tid >> 5;
  const int lane = tid & 31;
  const int j = lane & 15;
  const int h = lane >> 4;
  const int row0 = blockIdx.x * 128;  // samples ordered (b, to, n); tile stays in one (b,to)
  const int b = row0 / 14336;
  const int rem = row0 % 14336;
  const int to = rem >> 10;
  const int n0 = rem & 1023;

  v8f acc[NT];
  const v8f vzero = {0.f, 0.f, 0.f, 0.f, 0.f, 0.f, 0.f, 0.f};
#pragma unroll
  for (int i = 0; i < NT; ++i) acc[i] = vzero;

  for (int kc = 0; kc < 768; kc += 32) {
    const int dt = kc >> 8;
    const int c0 = kc & 255;
    __syncthreads();
    const float* xbase = xp + (size_t)((b * 16 + to + dt) * 1024 + n0) * 256 + c0;
    for (int idx = tid; idx < 4096; idx += 256) {
      const int r = idx >> 5, c = idx & 31;
      At[idx] = xbase[(size_t)r * 256 + c];
    }
    for (int idx = tid; idx < 32 * N; idx += 256) {
      const int kx = idx / N, n = idx % N;
      Bt[(((kx >> 1) * N) + n) * 2 + (kx & 1)] = wt[((size_t)n * 256 + c0 + kx) * 3 + dt];
    }
    __syncthreads();
#pragma unroll
    for (int kx = 0; kx < 32; kx += 4) {
      const v2f a = *(const v2f*)&At[(wave * 16 + j) * 32 + kx + 2 * h];
      const int p = (kx + 2 * h) >> 1;
#pragma unroll
      for (int nt = 0; nt < NT; ++nt) {
        const v2f bf = *(const v2f*)&Bt[(p * N + nt * 16 + j) * 2];
        acc[nt] = __builtin_amdgcn_wmma_f32_16x16x4_f32(false, a, false, bf, (short)0,
                                                        acc[nt], false, false);
      }
    }
  }
  __syncthreads();

  float* Wst = lds;
#pragma unroll
  for (int nt = 0; nt < NT; ++nt) {
    const int ch = nt * 16 + j;
    const float bv = bt[ch];
    float s = 0.f, q = 0.f;
#pragma unroll
    for (int v = 0; v < 8; ++v) {
      const float z = acc[nt][v] + bv;
      y[(size_t)(row0 + wave * 16 + h * 8 + v) * N + ch] = z;
      s += z;
      q += z * z;
    }
    s += __shfl_xor(s, 16, 32);
    q += __shfl_xor(q, 16, 32);
    if (h == 0) {
      Wst[(wave * N + ch) * 2 + 0] = s;
      Wst[(wave * N + ch) * 2 + 1] = q;
    }
  }
  __syncthreads();
  for (int ch = tid; ch < N; ch += 256) {
    float s = 0.f, q = 0.f;
#pragma unroll
    for (int w = 0; w < 8; ++w) {
      s += Wst[(w * N + ch) * 2 + 0];
      q += Wst[(w * N + ch) * 2 + 1];
    }
    statsP[((size_t)blockIdx.x * N + ch) * 2 + 0] = s;
    statsP[((size_t)blockIdx.x * N + ch) * 2 + 1] = q;
  }
}

__global__ __launch_bounds__(256) void bn_relu_inplace_kernel(float* __restrict__ y,
                                                              const float* __restrict__ sc,
                                                              const float* __restrict__ sh) {
  const int i = blockIdx.x * 256 + threadIdx.x;
  const int ch = i & 255;
  y[i] = fmaxf(fmaf(y[i], sc[ch], sh[ch]), 0.f);
}

// ---------------------------------------------------------------------------
extern "C" void kernel_launch(void* const* d_in, const int* in_sizes, int n_in,
                              void* d_out, int out_size, void* d_ws, size_t ws_size,
                              hipStream_t stream) {
  (void)in_sizes; (void)n_in; (void)out_size; (void)ws_size;
  const float* pts = (const float*)d_in[0];
  const float* w0 = (const float*)d_in[1];
  const float* b0 = (const float*)d_in[2];
  const float* g0 = (const float*)d_in[3];
  const float* be0 = (const float*)d_in[4];
  const float* w1 = (const float*)d_in[5];
  const float* b1 = (const float*)d_in[6];
  const float* g1 = (const float*)d_in[7];
  const float* be1 = (const float*)d_in[8];
  const float* w2 = (const float*)d_in[9];
  const float* b2 = (const float*)d_in[10];
  const float* g2 = (const float*)d_in[11];
  const float* be2 = (const float*)d_in[12];
  const float* wt = (const float*)d_in[13];
  const float* bt = (const float*)d_in[14];
  const float* gt = (const float*)d_in[15];
  const float* bet = (const float*)d_in[16];
  float* out = (float*)d_out;

  char* wsb = (char*)d_ws;
  size_t off = 0;
  auto alloc = [&](size_t nfloats) -> float* {
    float* p = (float*)(wsb + off);
    off += ((nfloats * 4 + 255) / 256) * 256;
    return p;
  };
  float* cur = alloc(6144);
  float* pp = alloc(1572864);          // (b,t,n,16,3)
  float* pf = alloc(1572864);
  float* z0 = alloc(33554432ull);      // 524288 x 64
  float* z1 = alloc(67108864ull);      // 524288 x 128
  float* gmax = alloc(8388608ull);     // (b,t,n) x 256
  float* gmin = alloc(8388608ull);
  float* xpool = alloc(8388608ull);
  float* st0 = alloc(2048ull * 64 * 2);
  float* st1 = alloc(4096ull * 128 * 2);
  float* st2 = alloc(4096ull * 256 * 2);
  float* stt = alloc(224ull * 256 * 2);
  float* sc0 = alloc(64); float* sh0 = alloc(64);
  float* sc1 = alloc(128); float* sh1 = alloc(128);
  float* sc2 = alloc(256); float* sh2 = alloc(256);
  float* sct = alloc(256); float* sht = alloc(256);

  init_cur_kernel<<<24, 256, 0, stream>>>(pts, cur);
  for (int t = 0; t < 16; ++t)
    knn_step_kernel<<<dim3(4, 2), 256, 0, stream>>>(pts, cur, pp, pf, t);

  layer0_kernel<<<2048, 256, 0, stream>>>(pp, pf, w0, b0, z0, st0);
  finalize_stats_kernel<<<64, 256, 0, stream>>>(st0, 2048, 64, 1.f / 524288.f, g0, be0, sc0, sh0);

  gemm_bn_kernel<64, 128, false><<<4096, 256, 0, stream>>>(z0, sc0, sh0, w1, b1, z1,
                                                           nullptr, nullptr, st1);
  finalize_stats_kernel<<<128, 256, 0, stream>>>(st1, 4096, 128, 1.f / 524288.f, g1, be1, sc1, sh1);

  gemm_bn_kernel<128, 256, true><<<4096, 256, 0, stream>>>(z1, sc1, sh1, w2, b2, nullptr,
                                                           gmax, gmin, st2);
  finalize_stats_kernel<<<256, 256, 0, stream>>>(st2, 4096, 256, 1.f / 524288.f, g2, be2, sc2, sh2);

  pool_bn_kernel<<<32768, 256, 0, stream>>>(gmax, gmin, sc2, sh2, xpool);

  tconv_kernel<<<224, 256, 0, stream>>>(xpool, wt, bt, out, stt);
  finalize_stats_kernel<<<256, 256, 0, stream>>>(stt, 224, 256, 1.f / 28672.f, gt, bet, sct, sht);
  bn_relu_inplace_kernel<<<28672, 256, 0, stream>>>(out, sct, sht);
}